// MolecularGraphEncoder_71717363909161
// MI455X (gfx1250) — compile-verified
//
#include <hip/hip_runtime.h>

#define TPB 256
// LDS layout for W panel: K-pair interleaved. Element (k,c) lives at
// lds[(k>>1)*PSTRIDE + 2*c + (k&1)], c in [0,64). PSTRIDE % 64 == 32 in
// floats => the two lane-halves (pair p vs p+1) hit disjoint bank halves.
#define PSTRIDE 160

typedef float v2f __attribute__((ext_vector_type(2)));
typedef float v8f __attribute__((ext_vector_type(8)));

__global__ void k_fill(float* __restrict__ p, float v, long long n) {
  long long i = (long long)blockIdx.x * blockDim.x + threadIdx.x;
  long long st = (long long)gridDim.x * blockDim.x;
  for (; i < n; i += st) p[i] = v;
}

__global__ void k_edge_w(const int* __restrict__ bond_ids,
                         const float* __restrict__ bond_emb,
                         float* __restrict__ ew, int E, int BD) {
  int e = blockIdx.x * blockDim.x + threadIdx.x;
  if (e >= E) return;
  const float* r = bond_emb + (long long)bond_ids[e] * BD;
  float s = 0.f;
  for (int i = 0; i < BD; ++i) s += r[i];
  ew[e] = s * (1.0f / (float)BD);
}

__global__ void k_deg_acc(const int* __restrict__ dst, const float* __restrict__ ew,
                          float* __restrict__ deg, int E) {
  int e = blockIdx.x * blockDim.x + threadIdx.x;
  if (e >= E) return;
  atomicAdd(&deg[dst[e]], ew[e]);
}

__global__ void k_deg_fin(float* __restrict__ deg_to_invdeg, float* __restrict__ dinv, int N) {
  int n = blockIdx.x * blockDim.x + threadIdx.x;
  if (n >= N) return;
  float d = deg_to_invdeg[n];
  dinv[n] = rsqrtf(d);
  deg_to_invdeg[n] = 1.0f / d;
}

__global__ void k_norm(const int* __restrict__ src, const int* __restrict__ dst,
                       const float* __restrict__ dinv, float* __restrict__ nrm, int E) {
  int e = blockIdx.x * blockDim.x + threadIdx.x;
  if (e >= E) return;
  nrm[e] = dinv[src[e]] * nrm[e] * dinv[dst[e]];
}

__global__ void k_embed(const int* __restrict__ ids, const float* __restrict__ emb,
                        float* __restrict__ X, int N, int D) {
  long long i = (long long)blockIdx.x * blockDim.x + threadIdx.x;
  if (i >= (long long)N * D) return;
  int n = (int)(i / D);
  int f = (int)(i - (long long)n * D);
  X[i] = emb[(long long)ids[n] * D + f];
}

// H[N,F] = X[N,K] @ W[K,F], fp32 via V_WMMA_F32_16X16X4_F32.
// Block = 8 waves, block tile = 128 rows x 64 cols.
// W panel (K x 64) staged in LDS with K-pair interleaving so each lane's
// B fragment (k, k+1) is one aligned ds_load_b64 straight into an even
// VGPR pair (no repack moves). Each wave: one 16-row strip, 4 accumulators,
// so one A fragment feeds 4 WMMAs per k-step.
__global__ void k_gemm_wmma(const float* __restrict__ X, const float* __restrict__ W,
                            float* __restrict__ H, int N, int K, int F, int colPanels) {
  extern __shared__ float lds[];

  int panel    = blockIdx.x % colPanels;
  int rowBlock = blockIdx.x / colPanels;
  int colBase  = panel << 6;              // 64 cols per panel
  int tid      = threadIdx.x;

  // ---- stage W[K, colBase:colBase+64] into LDS, K-pair interleaved ----
  // index -> (pair p, float4-column c4); read rows 2p and 2p+1, write two
  // interleaved float4s: (a.x,b.x,a.y,b.y) (a.z,b.z,a.w,b.w)
  int nVec = (K >> 1) << 4;               // (K/2) * 16
  for (int idx = tid; idx < nVec; idx += TPB) {
    int p  = idx >> 4;
    int c4 = idx & 15;
    const float4 a = *(const float4*)(W + (long long)(2 * p) * F + colBase + (c4 << 2));
    const float4 b = *(const float4*)(W + (long long)(2 * p + 1) * F + colBase + (c4 << 2));
    float4* o = (float4*)&lds[p * PSTRIDE + (c4 << 3)];
    o[0] = make_float4(a.x, b.x, a.y, b.y);
    o[1] = make_float4(a.z, b.z, a.w, b.w);
  }
  __syncthreads();

  int wave = tid >> 5;
  int lane = tid & 31;
  int l15  = lane & 15;
  int hi   = lane >> 4;

  int rowBase = (rowBlock << 7) + (wave << 4);
  int arow = rowBase + l15;
  if (arow > N - 1) arow = N - 1;         // clamp; stores guarded below
  const float* xptr  = X + (long long)arow * K + (hi << 1);
  const float* bbase = lds + hi * PSTRIDE + (l15 << 1);

  v8f acc[4] = {};
  for (int k0 = 0; k0 < K; k0 += 4) {
    v2f a = *(const v2f*)(xptr + k0);
    __builtin_prefetch(xptr + k0 + 16, 0, 1);
    const float* b0 = bbase + (k0 >> 1) * PSTRIDE;
#pragma unroll
    for (int j = 0; j < 4; ++j) {
      v2f b = *(const v2f*)(b0 + (j << 5));   // cols advance 16 -> 32 floats
      acc[j] = __builtin_amdgcn_wmma_f32_16x16x4_f32(false, a, false, b,
                                                     (short)0, acc[j], false, false);
    }
  }

#pragma unroll
  for (int j = 0; j < 4; ++j) {
#pragma unroll
    for (int v = 0; v < 8; ++v) {
      int r = rowBase + (hi << 3) + v;
      if (r < N) H[(long long)r * F + colBase + (j << 4) + l15] = acc[j][v];
    }
  }
}

__global__ void k_scatter(const float* __restrict__ H, const float* __restrict__ nrm,
                          const int* __restrict__ src, const int* __restrict__ dst,
                          float* __restrict__ agg, int E, int F, int lf) {
  long long total = (long long)E * F;
  long long st = (long long)gridDim.x * blockDim.x;
  for (long long i = (long long)blockIdx.x * blockDim.x + threadIdx.x; i < total; i += st) {
    int e = (int)(i >> lf);
    int f = (int)(i & (F - 1));
    float v = H[((long long)src[e] << lf) + f] * nrm[e];
    atomicAdd(&agg[((long long)dst[e] << lf) + f], v);
  }
}

__global__ void k_finalize(const float* __restrict__ agg, const float* __restrict__ H,
                           const float* __restrict__ invdeg, const float* __restrict__ bias,
                           float* __restrict__ Y, int N, int F, int lf, int relu) {
  long long i = (long long)blockIdx.x * blockDim.x + threadIdx.x;
  if (i >= (long long)N * F) return;
  int n = (int)(i >> lf);
  int f = (int)(i & (F - 1));
  float v = agg[i] + H[i] * invdeg[n] + bias[f];
  if (relu) v = fmaxf(v, 0.0f);
  Y[i] = v;
}

__global__ void k_pool_cnt(const int* __restrict__ batch, float* __restrict__ cnt, int N) {
  int n = blockIdx.x * blockDim.x + threadIdx.x;
  if (n >= N) return;
  atomicAdd(&cnt[batch[n]], 1.0f);
}

__global__ void k_pool_sum(const float* __restrict__ X, const int* __restrict__ batch,
                           float* __restrict__ out, int N, int F, int lf) {
  long long i = (long long)blockIdx.x * blockDim.x + threadIdx.x;
  if (i >= (long long)N * F) return;
  int n = (int)(i >> lf);
  int f = (int)(i & (F - 1));
  atomicAdd(&out[((long long)batch[n] << lf) + f], X[i]);
}

__global__ void k_pool_div(float* __restrict__ out, const float* __restrict__ cnt,
                           int total, int lf) {
  int i = blockIdx.x * blockDim.x + threadIdx.x;
  if (i >= total) return;
  int g = i >> lf;
  out[i] /= fmaxf(cnt[g], 1.0f);
}

extern "C" void kernel_launch(void* const* d_in, const int* in_sizes, int n_in,
                              void* d_out, int out_size, void* d_ws, size_t ws_size,
                              hipStream_t stream) {
  (void)n_in; (void)ws_size;
  const int*   atom_ids = (const int*)d_in[0];
  const int*   bond_ids = (const int*)d_in[1];
  const int*   edge_idx = (const int*)d_in[2];
  const int*   batch    = (const int*)d_in[3];
  // d_in[4] = num_graphs (device scalar); G derived from out_size instead.
  const float* atom_emb = (const float*)d_in[5];
  const float* bond_emb = (const float*)d_in[6];
  const float* W1 = (const float*)d_in[7];  const float* b1 = (const float*)d_in[8];
  const float* W2 = (const float*)d_in[9];  const float* b2 = (const float*)d_in[10];
  const float* W3 = (const float*)d_in[11]; const float* b3 = (const float*)d_in[12];

  const int N  = in_sizes[0];
  const int E  = in_sizes[1];
  const int F1 = in_sizes[8];          // 128
  const int F2 = in_sizes[10];         // 256
  const int F3 = in_sizes[12];         // 512
  const int K1 = in_sizes[7] / F1;     // 64
  const int BD = 32;                   // bond embedding width (mean over axis 1)
  const int G  = out_size / F3;        // 256

  const int* src = edge_idx;
  const int* dst = edge_idx + E;

  // workspace carve-up (floats)
  float* w      = (float*)d_ws;
  float* nrm    = w;                       // E   (edge weight, then norm in-place)
  float* dinv   = nrm + E;                 // N
  float* invdeg = dinv + N;                // N   (deg accumulator, then 1/deg)
  float* cnt    = invdeg + N;              // G
  long long nb  = (long long)N * F3;
  float* bufX   = cnt + G;                 // N*F3 (current activations)
  float* bufH   = bufX + nb;               // N*F3 (x @ W)
  float* bufAgg = bufH + nb;               // N*F3 (scatter accumulator)

  // 1) edge weights
  k_edge_w<<<(E + TPB - 1) / TPB, TPB, 0, stream>>>(bond_ids, bond_emb, nrm, E, BD);
  // 2) degree (self-loop = 1) -> dinv, invdeg
  k_fill<<<(N + TPB - 1) / TPB, TPB, 0, stream>>>(invdeg, 1.0f, N);
  k_deg_acc<<<(E + TPB - 1) / TPB, TPB, 0, stream>>>(dst, nrm, invdeg, E);
  k_deg_fin<<<(N + TPB - 1) / TPB, TPB, 0, stream>>>(invdeg, dinv, N);
  // 3) norm[e] = dinv[src]*ew*dinv[dst]
  k_norm<<<(E + TPB - 1) / TPB, TPB, 0, stream>>>(src, dst, dinv, nrm, E);
  // 4) node embedding gather
  {
    long long tot = (long long)N * K1;
    k_embed<<<(int)((tot + TPB - 1) / TPB), TPB, 0, stream>>>(atom_ids, atom_emb, bufX, N, K1);
  }

  // 5) three GCN layers
  const float* Ws[3] = {W1, W2, W3};
  const float* bs[3] = {b1, b2, b3};
  int Ks[3] = {K1, F1, F2};
  int Fs[3] = {F1, F2, F3};
  for (int L = 0; L < 3; ++L) {
    int K = Ks[L], F = Fs[L];
    int lf = 0; while ((1 << lf) < F) ++lf;
    int rowBlocks = (N + 127) / 128;
    int colPanels = F >> 6;                       // 64 cols per panel
    int ldsBytes  = (K >> 1) * PSTRIDE * (int)sizeof(float);
    k_gemm_wmma<<<rowBlocks * colPanels, TPB, ldsBytes, stream>>>(
        bufX, Ws[L], bufH, N, K, F, colPanels);
    long long nf = (long long)N * F;
    k_fill<<<2048, TPB, 0, stream>>>(bufAgg, 0.0f, nf);
    k_scatter<<<4096, TPB, 0, stream>>>(bufH, nrm, src, dst, bufAgg, E, F, lf);
    k_finalize<<<(int)((nf + TPB - 1) / TPB), TPB, 0, stream>>>(
        bufAgg, bufH, invdeg, bs[L], bufX, N, F, lf, (L < 2) ? 1 : 0);
  }

  // 6) per-graph mean pool
  {
    int lf3 = 0; while ((1 << lf3) < F3) ++lf3;
    k_fill<<<64, TPB, 0, stream>>>((float*)d_out, 0.0f, out_size);
    k_fill<<<1, TPB, 0, stream>>>(cnt, 0.0f, G);
    k_pool_cnt<<<(N + TPB - 1) / TPB, TPB, 0, stream>>>(batch, cnt, N);
    long long nf = (long long)N * F3;
    k_pool_sum<<<(int)((nf + TPB - 1) / TPB), TPB, 0, stream>>>(bufX, batch,
                                                                (float*)d_out, N, F3, lf3);
    k_pool_div<<<(out_size + TPB - 1) / TPB, TPB, 0, stream>>>((float*)d_out, cnt,
                                                               out_size, lf3);
  }
}